// StackChamferDistance_18846316495074
// MI455X (gfx1250) — compile-verified
//
#include <hip/hip_runtime.h>

typedef __attribute__((ext_vector_type(2))) float v2f;
typedef __attribute__((ext_vector_type(4))) float v4f;
typedef __attribute__((ext_vector_type(8))) float v8f;

#define NPER  2048
#define NB    16
#define TILES (NPER / 16)   // 128 y-tiles per segment
#define RGPB  (NPER / 16)   // 128 row-groups per batch

// One wave owns 16 rows of X in one segment and streams all y-tiles of the
// same segment through V_WMMA_F32_16X16X4_F32.
//
// The spare K slot (K=3) carries the yy bias:  A[M][3] = 1, B[3][N] = ||y_N||^2
// The C accumulator carries the xx bias:       C[M][N] = ||x_M||^2 (loop-invariant)
// So one WMMA produces the full squared-distance tile:
//   D = xx[M] + yy[N] - 2*x_M . y_N
// and the loop body is just: load -> yy -> 2 cndmask -> WMMA -> 8 min.
__global__ __launch_bounds__(256) void chamfer_rowmin_wmma(
    const float* __restrict__ X, const float* __restrict__ Y,
    float* __restrict__ dist)
{
  const int lane = threadIdx.x & 31;
  const int wave = (blockIdx.x * 256 + threadIdx.x) >> 5;
  const int b    = wave >> 7;          // wave / RGPB
  const int rg   = wave & (RGPB - 1);
  const int rowbase = b * NPER + rg * 16;

  const int  lp = lane & 15;           // point index within 16-wide tile
  const bool hi = lane >= 16;          // high half-wave supplies K=2,3

  // ---- A tile: lane<16 -> (-2x0, -2x1); lane>=16 -> (-2x2, 1.0) ----
  const float* xp = X + (size_t)(rowbase + lp) * 3;
  const float x0 = xp[0], x1 = xp[1], x2 = xp[2];
  const float xx = x0 * x0 + x1 * x1 + x2 * x2;

  v2f a;
  a.x = hi ? (-2.0f * x2) : (-2.0f * x0);
  a.y = hi ? 1.0f         : (-2.0f * x1);

  // cxx[r] = ||x_row||^2 for the row (r + 8*hi) this lane's C-element r maps
  // to. Row m's norm lives in lanes m and m+16; pull with full-wave shuffles.
  v8f cxx;
#pragma unroll
  for (int r = 0; r < 8; ++r)
    cxx[r] = __shfl(xx, hi ? (24 + r) : r, 32);

  v8f rmin;
#pragma unroll
  for (int r = 0; r < 8; ++r) rmin[r] = 3.0e38f;

  const float* ybase = Y + ((size_t)b * NPER + lp) * 3;

#pragma unroll 4
  for (int t = 0; t < TILES; ++t) {
    // ---- B tile: lane<16 -> (y0, y1); lane>=16 -> (y2, yy) ----
    const float* yp = ybase + (size_t)t * 48;   // 16 points * 12 B
    const float y0 = yp[0], y1 = yp[1], y2 = yp[2];
    const float yy = y0 * y0 + y1 * y1 + y2 * y2;

    v2f bb;
    bb.x = hi ? y2 : y0;
    bb.y = hi ? yy : y1;

    v8f d = __builtin_amdgcn_wmma_f32_16x16x4_f32(
        /*neg_a=*/false, a, /*neg_b=*/false, bb,
        /*c_mod=*/(short)0, cxx, /*reuse_a=*/false, /*reuse_b=*/false);

#pragma unroll
    for (int r = 0; r < 8; ++r) rmin[r] = fminf(rmin[r], d[r]);
  }

  // Row-min across the 16 columns: xor masks < 16 keep the two half-waves
  // (rows r vs rows r+8) independent.
#pragma unroll
  for (int m = 1; m <= 8; m <<= 1) {
#pragma unroll
    for (int r = 0; r < 8; ++r)
      rmin[r] = fminf(rmin[r], __shfl_xor(rmin[r], m, 32));
  }
  // clamp(d, 0) commutes with min
#pragma unroll
  for (int r = 0; r < 8; ++r) rmin[r] = fmaxf(rmin[r], 0.0f);

  // Lane 0 -> rows [rowbase, rowbase+8), lane 16 -> rows [rowbase+8, rowbase+16)
  if (lp == 0) {
    float* o = dist + rowbase + (hi ? 8 : 0);
    v4f lo = { rmin[0], rmin[1], rmin[2], rmin[3] };
    v4f h4 = { rmin[4], rmin[5], rmin[6], rmin[7] };
    *reinterpret_cast<v4f*>(o)     = lo;
    *reinterpret_cast<v4f*>(o + 4) = h4;
  }
}

// deg_x / deg_y: equal-size segments of 2048 points each.
__global__ void write_deg_kernel(float* __restrict__ o)
{
  const int i = threadIdx.x;
  if (i < 2 * NB) o[i] = (float)NPER;
}

extern "C" void kernel_launch(void* const* d_in, const int* in_sizes, int n_in,
                              void* d_out, int out_size, void* d_ws, size_t ws_size,
                              hipStream_t stream)
{
  (void)in_sizes; (void)n_in; (void)out_size; (void)d_ws; (void)ws_size;

  const float* x = (const float*)d_in[0];
  const float* y = (const float*)d_in[1];
  float* out = (float*)d_out;

  const int waves  = (NB * NPER) / 16;   // 2048
  const int blocks = waves / 8;          // 256 blocks of 8 waves

  // dist_x: nearest y for each x
  chamfer_rowmin_wmma<<<blocks, 256, 0, stream>>>(x, y, out);
  // dist_y: symmetric problem with roles swapped
  chamfer_rowmin_wmma<<<blocks, 256, 0, stream>>>(y, x, out + NB * NPER);
  // deg_x, deg_y
  write_deg_kernel<<<1, 64, 0, stream>>>(out + 2 * NB * NPER);
}